// GraphNet_17368847745176
// MI455X (gfx1250) — compile-verified
//
#include <hip/hip_runtime.h>
#include <hip/hip_bf16.h>

// Problem constants (match reference)
#define G_   128
#define P_   512
#define K_   16
#define FEAT 128
#define NPTS (G_ * P_)   // 65536

typedef __attribute__((ext_vector_type(2))) float v2f;
typedef __attribute__((ext_vector_type(8))) float v8f;

// ---------------------------------------------------------------------------
// Kernel 1: KNN per graph. One block per graph, one thread per target point.
// Integer 2-D squared distance (graph id column is constant within a graph).
// Maintains a sorted ascending (d2, idx) list of size 16 in registers; strict
// comparisons reproduce lax.top_k's lower-index tie-breaking.
// ---------------------------------------------------------------------------
__global__ __launch_bounds__(P_) void knn_kernel(const int* __restrict__ coo,
                                                 int* __restrict__ knnOut) {
  __shared__ int sx[P_];
  __shared__ int sy[P_];
  const int g = blockIdx.x;
  const int p = threadIdx.x;
  const int n = (g << 9) + p;
  const int mx = coo[n * 3 + 0];
  const int my = coo[n * 3 + 1];
  sx[p] = mx;
  sy[p] = my;
  __syncthreads();

  int bd[K_], bi[K_];
#pragma unroll
  for (int j = 0; j < K_; ++j) { bd[j] = 0x7fffffff; bi[j] = 0; }

  for (int j = 0; j < P_; ++j) {
    if (j == p) continue;                 // loop = False (self excluded)
    const int dx = sx[j] - mx;
    const int dy = sy[j] - my;
    const int d2 = dx * dx + dy * dy;
    if (d2 < bd[K_ - 1]) {                // strict: ties keep earlier index
#pragma unroll
      for (int q = K_ - 1; q > 0; --q) {
        if (bd[q - 1] > d2)      { bd[q] = bd[q - 1]; bi[q] = bi[q - 1]; }
        else if (bd[q] > d2)     { bd[q] = d2;        bi[q] = j;         }
      }
      if (bd[0] > d2)            { bd[0] = d2;        bi[0] = j;         }
    }
  }

  int* o = knnOut + (size_t)n * K_;
#pragma unroll
  for (int j = 0; j < K_; ++j) o[j] = bi[j];
}

// ---------------------------------------------------------------------------
// Kernel 2: neighbor mean. One block per point, one thread per input channel.
// Gathers are coalesced over channels and hit L2 (per-graph activations are
// only a few hundred KB).
// ---------------------------------------------------------------------------
__global__ __launch_bounds__(FEAT) void mean_kernel(const float* __restrict__ hin,
                                                    const int* __restrict__ knn,
                                                    float* __restrict__ meanOut,
                                                    int nin) {
  const int p = blockIdx.x;              // global point index
  const int g = p >> 9;                  // / P_
  const int c = threadIdx.x;
  const int* mi = knn + (size_t)p * K_;
  const float* base = hin + (size_t)(g << 9) * nin;
  float s = 0.f;
#pragma unroll
  for (int j = 0; j < K_; ++j) {
    const int nb = mi[j];                // per-graph local neighbor index
    s += base[(size_t)nb * nin + c];
  }
  meanOut[(size_t)p * nin + c] = s * (1.0f / K_);
}

// ---------------------------------------------------------------------------
// Kernel 3: SAGE layer GEMM on the WMMA pipe.
//   out = relu(mean @ wl + h @ wr + b), out tile 16x16 per wave.
// 8 waves / block share an LDS-staged 16-column slice of wl and wr.
// A fragment (16x4 f32): lane<16 -> row m0+lane, K=k0,k0+1 (float2 load);
//                        lane>=16 -> row m0+lane-16, K=k0+2,k0+3.
// B fragment (4x16 f32): mirrored over N.  C/D: VGPR r holds rows m0+r (lanes
// 0-15) and m0+8+r (lanes 16-31), col n0+(lane&15).
// ---------------------------------------------------------------------------
template <int NIN, int NOUT>
__global__ __launch_bounds__(256) void sage_gemm(const float* __restrict__ meanIn,
                                                 const float* __restrict__ hIn,
                                                 const float* __restrict__ wl,
                                                 const float* __restrict__ wr,
                                                 const float* __restrict__ bias,
                                                 float* __restrict__ out) {
  __shared__ float sWl[NIN * 16];
  __shared__ float sWr[NIN * 16];
  const int tid = threadIdx.x;
  const int n0 = blockIdx.y * 16;

  for (int e = tid; e < NIN * 16; e += 256) {
    const int k = e >> 4;
    const int nn = e & 15;
    sWl[e] = wl[k * NOUT + n0 + nn];
    sWr[e] = wr[k * NOUT + n0 + nn];
  }
  __syncthreads();

  const int lane = tid & 31;
  const int wave = tid >> 5;
  const int m0 = (blockIdx.x * 8 + wave) * 16;
  const int nn = lane & 15;
  const int khi = (lane >> 4) << 1;       // 0 for lanes 0-15, 2 for 16-31
  const int arow = m0 + nn;               // A row handled by this lane

  const float bv = bias[n0 + nn];
  v8f acc;
#pragma unroll
  for (int r = 0; r < 8; ++r) acc[r] = bv;

  const float* mrow = meanIn + (size_t)arow * NIN;
  const float* hrow = hIn + (size_t)arow * NIN;

  for (int k0 = 0; k0 < NIN; k0 += 4) {
    const int kk = k0 + khi;
    const v2f aM = *(const v2f*)(mrow + kk);   // 8B aligned: kk even, NIN even
    const v2f aH = *(const v2f*)(hrow + kk);
    v2f bL, bR;
    bL[0] = sWl[kk * 16 + nn];
    bL[1] = sWl[(kk + 1) * 16 + nn];
    bR[0] = sWr[kk * 16 + nn];
    bR[1] = sWr[(kk + 1) * 16 + nn];
    acc = __builtin_amdgcn_wmma_f32_16x16x4_f32(false, aM, false, bL,
                                                (short)0, acc, false, false);
    acc = __builtin_amdgcn_wmma_f32_16x16x4_f32(false, aH, false, bR,
                                                (short)0, acc, false, false);
  }

  const int rowBase = m0 + ((lane >> 4) << 3);
#pragma unroll
  for (int r = 0; r < 8; ++r) {
    const float v = acc[r] > 0.f ? acc[r] : 0.f;     // fused ReLU
    out[(size_t)(rowBase + r) * NOUT + n0 + nn] = v;
  }
}

// ---------------------------------------------------------------------------
// Kernel 4: global max pool over the 512 points of each graph + 32->32->3 MLP.
// One block (8 waves) per graph.
// ---------------------------------------------------------------------------
__global__ __launch_bounds__(256) void pool_mlp_kernel(const float* __restrict__ h,
                                                       const float* __restrict__ lw0,
                                                       const float* __restrict__ lb0,
                                                       const float* __restrict__ lw1,
                                                       const float* __restrict__ lb1,
                                                       float* __restrict__ out) {
  __shared__ float part[8][32];
  __shared__ float pooled[32];
  __shared__ float hid[32];
  const int g = blockIdx.x;
  const int tid = threadIdx.x;
  const int c = tid & 31;
  const int seg = tid >> 5;

  float m = -3.402823466e+38f;
  for (int p = seg; p < P_; p += 8)
    m = fmaxf(m, h[((size_t)(g << 9) + p) * 32 + c]);
  part[seg][c] = m;
  __syncthreads();

  if (tid < 32) {
    float mm = part[0][c];
#pragma unroll
    for (int s = 1; s < 8; ++s) mm = fmaxf(mm, part[s][c]);
    pooled[c] = mm;
  }
  __syncthreads();

  if (tid < 32) {
    float a = lb0[c];
#pragma unroll
    for (int i = 0; i < 32; ++i) a += pooled[i] * lw0[i * 32 + c];
    hid[c] = a > 0.f ? a : 0.f;
  }
  __syncthreads();

  if (tid < 3) {
    float a = lb1[tid];
#pragma unroll
    for (int i = 0; i < 32; ++i) a += hid[i] * lw1[i * 3 + tid];
    out[g * 3 + tid] = a;
  }
}

// ---------------------------------------------------------------------------
// Host-side launcher
// ---------------------------------------------------------------------------
extern "C" void kernel_launch(void* const* d_in, const int* in_sizes, int n_in,
                              void* d_out, int out_size, void* d_ws, size_t ws_size,
                              hipStream_t stream) {
  (void)in_sizes; (void)n_in; (void)out_size; (void)ws_size;

  const float* x   = (const float*)d_in[0];
  const int*   coo = (const int*)d_in[1];
  const float* wl0 = (const float*)d_in[2];
  const float* wr0 = (const float*)d_in[3];
  const float* b0  = (const float*)d_in[4];
  const float* wl1 = (const float*)d_in[5];
  const float* wr1 = (const float*)d_in[6];
  const float* b1  = (const float*)d_in[7];
  const float* wl2 = (const float*)d_in[8];
  const float* wr2 = (const float*)d_in[9];
  const float* b2  = (const float*)d_in[10];
  const float* lw0 = (const float*)d_in[11];
  const float* lb0 = (const float*)d_in[12];
  const float* lw1 = (const float*)d_in[13];
  const float* lb1 = (const float*)d_in[14];
  float* outp = (float*)d_out;

  // Workspace carve-up (bytes):
  //   [0,            4194304)  knn indices  G*P*K int
  //   [4194304,     37748736)  bufA: mean buffer, up to 65536x128 f32
  //   [37748736,    62914560)  bufB: activations, up to 65536x96 f32
  //   [62914560,    79691776)  bufC: activations, 65536x64 f32
  char* ws = (char*)d_ws;
  int*   knn  = (int*)ws;
  float* bufA = (float*)(ws + (size_t)4194304);
  float* bufB = (float*)(ws + (size_t)37748736);
  float* bufC = (float*)(ws + (size_t)62914560);

  // 1) KNN
  knn_kernel<<<G_, P_, 0, stream>>>(coo, knn);

  // 2) Layer 0: 128 -> 96
  mean_kernel<<<NPTS, 128, 0, stream>>>(x, knn, bufA, 128);
  sage_gemm<128, 96><<<dim3(NPTS / 16 / 8, 96 / 16), 256, 0, stream>>>(
      bufA, x, wl0, wr0, b0, bufB);

  // 3) Layer 1: 96 -> 64
  mean_kernel<<<NPTS, 96, 0, stream>>>(bufB, knn, bufA, 96);
  sage_gemm<96, 64><<<dim3(NPTS / 16 / 8, 64 / 16), 256, 0, stream>>>(
      bufA, bufB, wl1, wr1, b1, bufC);

  // 4) Layer 2: 64 -> 32
  mean_kernel<<<NPTS, 64, 0, stream>>>(bufC, knn, bufA, 64);
  sage_gemm<64, 32><<<dim3(NPTS / 16 / 8, 32 / 16), 256, 0, stream>>>(
      bufA, bufC, wl2, wr2, b2, bufB);

  // 5) Global max pool + MLP head
  pool_mlp_kernel<<<G_, 256, 0, stream>>>(bufB, lw0, lb0, lw1, lb1, outp);
}